// region_proposal_40638980555104
// MI455X (gfx1250) — compile-verified
//
#include <hip/hip_runtime.h>
#include <stdint.h>

// ---------------- problem constants (match reference) ----------------
#define N_ANCHORS   36864
#define NSORT       65536          // next pow2 >= N_ANCHORS for bitonic sort
#define PRE_NMS     6000
#define MASK_WORDS  188            // ceil(6000/32)
#define ROWS_PAD    6016           // 188*32, padded row count
#define POST_NMS    300
#define NMS_THRESH  0.7f
#define MIN_SIZE    16.0f
#define NEG_INF     (-1e9f)

#define SEG         2048           // keys per local-sort block (16KB LDS)

// ---------------- workspace layout (bytes, all 16B aligned) ----------------
#define OFF_BBOX    0u             // 36864*4 f32              = 589824
#define OFF_KEYS    589824u        // 65536 u64                = 524288
#define OFF_BBOXS   1114112u       // 6016*4 f32               = 96256
#define OFF_VALID   1210368u       // 6016 u32                 = 24064
#define OFF_MASK    1234432u       // 6016*188 u32             = 4524032
#define OFF_KEEP    5758464u       // 188 u32                  = 752

__device__ __forceinline__ uint32_t f32_sortable(float s) {
  uint32_t u = __float_as_uint(s);
  return (u & 0x80000000u) ? ~u : (u | 0x80000000u);
}

// async global->LDS 16B copy (per active lane), CDNA5 path
__device__ __forceinline__ void async_copy_b128(uint32_t lds_addr,
                                                unsigned long long gaddr) {
  asm volatile("global_load_async_to_lds_b128 %0, %1, off"
               :: "v"(lds_addr), "v"(gaddr) : "memory");
}
__device__ __forceinline__ void async_wait0() {
  asm volatile("s_wait_asynccnt 0" ::: "memory");
}

// ---------------------------------------------------------------------
// 1) decode + clamp + validity + pack sortable keys
// ---------------------------------------------------------------------
__global__ void k_decode(const float* __restrict__ anchors,
                         const float* __restrict__ cls,
                         const float* __restrict__ reg,
                         const int* __restrict__ img_w_p,
                         const int* __restrict__ img_h_p,
                         float* __restrict__ bbox,
                         unsigned long long* __restrict__ keys) {
  int i = blockIdx.x * blockDim.x + threadIdx.x;
  if (i >= NSORT) return;
  if (i >= N_ANCHORS) { keys[i] = 0ull; return; }  // pad sinks to bottom

  float a0 = anchors[i * 4 + 0], a1 = anchors[i * 4 + 1];
  float a2 = anchors[i * 4 + 2], a3 = anchors[i * 4 + 3];
  float r0 = reg[i * 4 + 0], r1 = reg[i * 4 + 1];
  float r2 = reg[i * 4 + 2], r3 = reg[i * 4 + 3];

  float w  = a2 - a0, h = a3 - a1;
  float cx = a0 + 0.5f * w, cy = a1 + 0.5f * h;
  float pcx = r0 * w + cx, pcy = r1 * h + cy;
  float pw  = expf(r2) * w, ph = expf(r3) * h;
  float x0 = pcx - 0.5f * pw, y0 = pcy - 0.5f * ph;
  float x1 = pcx + 0.5f * pw, y1 = pcy + 0.5f * ph;

  // reference clamps cols 0,2 with img_h and cols 1,3 with img_w
  float H = (float)(*img_h_p), W = (float)(*img_w_p);
  x0 = fminf(fmaxf(x0, 0.0f), H);
  x1 = fminf(fmaxf(x1, 0.0f), H);
  y0 = fminf(fmaxf(y0, 0.0f), W);
  y1 = fminf(fmaxf(y1, 0.0f), W);

  bbox[i * 4 + 0] = x0; bbox[i * 4 + 1] = y0;
  bbox[i * 4 + 2] = x1; bbox[i * 4 + 3] = y1;

  bool valid = (x1 - x0 >= MIN_SIZE) && (y1 - y0 >= MIN_SIZE);
  float s = valid ? cls[i] : NEG_INF;
  uint32_t ks = f32_sortable(s);
  // descending sort => high score first; tie-break: smaller index first
  keys[i] = ((unsigned long long)ks << 32) |
            (unsigned long long)(0xFFFFFFFFu - (uint32_t)i);
}

// ---------------------------------------------------------------------
// 2a) fused local bitonic sort: all (k,j) phases with k<=SEG in LDS.
//     2048 keys/block, 1024 threads, async LDS fill.
// ---------------------------------------------------------------------
__global__ void __launch_bounds__(1024) k_sort_local(unsigned long long* __restrict__ keys) {
  __shared__ __align__(16) unsigned long long s[SEG];
  const int t    = threadIdx.x;                 // 0..1023
  const int base = blockIdx.x * SEG;

  // async fill: each thread copies 16B (two keys)
  async_copy_b128((uint32_t)(uintptr_t)(&s[2 * t]),
                  (unsigned long long)(uintptr_t)(keys + base + 2 * t));
  async_wait0();
  __syncthreads();

  for (int k = 2; k <= SEG; k <<= 1) {
    for (int j = k >> 1; j > 0; j >>= 1) {
      int i   = ((t & ~(j - 1)) << 1) | (t & (j - 1));
      int ixj = i | j;
      bool desc = (((base + i) & k) == 0);      // global-index direction
      unsigned long long a = s[i], b = s[ixj];
      bool sw = desc ? (a < b) : (a > b);
      if (sw) { s[i] = b; s[ixj] = a; }
      __syncthreads();
    }
  }
  keys[base + t]        = s[t];
  keys[base + t + 1024] = s[t + 1024];
}

// ---------------------------------------------------------------------
// 2b) one global bitonic pass (only for strides j >= SEG)
// ---------------------------------------------------------------------
__global__ void k_bitonic_global(unsigned long long* __restrict__ keys, int k, int j) {
  int i = blockIdx.x * blockDim.x + threadIdx.x;
  int ixj = i ^ j;
  if (ixj <= i) return;
  unsigned long long a = keys[i], b = keys[ixj];
  bool desc = ((i & k) == 0);
  bool sw = desc ? (a < b) : (a > b);
  if (sw) { keys[i] = b; keys[ixj] = a; }
}

// ---------------------------------------------------------------------
// 2c) fused local merge for one k: all strides j<=SEG/2 in LDS
// ---------------------------------------------------------------------
__global__ void __launch_bounds__(1024) k_merge_local(unsigned long long* __restrict__ keys, int k) {
  __shared__ __align__(16) unsigned long long s[SEG];
  const int t    = threadIdx.x;
  const int base = blockIdx.x * SEG;

  async_copy_b128((uint32_t)(uintptr_t)(&s[2 * t]),
                  (unsigned long long)(uintptr_t)(keys + base + 2 * t));
  async_wait0();
  __syncthreads();

  const bool desc = ((base & k) == 0);          // uniform: k >= 2*SEG here
  for (int j = SEG >> 1; j > 0; j >>= 1) {
    int i   = ((t & ~(j - 1)) << 1) | (t & (j - 1));
    int ixj = i | j;
    unsigned long long a = s[i], b = s[ixj];
    bool sw = desc ? (a < b) : (a > b);
    if (sw) { s[i] = b; s[ixj] = a; }
    __syncthreads();
  }
  keys[base + t]        = s[t];
  keys[base + t + 1024] = s[t + 1024];
}

// ---------------------------------------------------------------------
// 3) gather top-6016 boxes (rows >= 6000 zero-padded) + validity
// ---------------------------------------------------------------------
__global__ void k_gather(const unsigned long long* __restrict__ keys,
                         const float* __restrict__ bbox,
                         float* __restrict__ bbox_s,
                         uint32_t* __restrict__ valid_s) {
  int i = blockIdx.x * blockDim.x + threadIdx.x;
  if (i >= ROWS_PAD) return;
  if (i >= PRE_NMS) {
    bbox_s[i * 4 + 0] = 0.0f; bbox_s[i * 4 + 1] = 0.0f;
    bbox_s[i * 4 + 2] = 0.0f; bbox_s[i * 4 + 3] = 0.0f;
    valid_s[i] = 0u;
    return;
  }
  uint32_t idx = 0xFFFFFFFFu - (uint32_t)(keys[i] & 0xFFFFFFFFull);
  float x0 = bbox[(size_t)idx * 4 + 0], y0 = bbox[(size_t)idx * 4 + 1];
  float x1 = bbox[(size_t)idx * 4 + 2], y1 = bbox[(size_t)idx * 4 + 3];
  bbox_s[i * 4 + 0] = x0; bbox_s[i * 4 + 1] = y0;
  bbox_s[i * 4 + 2] = x1; bbox_s[i * 4 + 3] = y1;
  valid_s[i] = ((x1 - x0 >= MIN_SIZE) && (y1 - y0 >= MIN_SIZE)) ? 1u : 0u;
}

// ---------------------------------------------------------------------
// 4) IoU suppression bitmask: 8 waves/block share one async-staged
//    column tile; each wave handles one 32-row group vs the 32 columns.
// ---------------------------------------------------------------------
__global__ void __launch_bounds__(256) k_mask(const float* __restrict__ bbox_s,
                                              uint32_t* __restrict__ mask) {
  __shared__ __align__(16) float cb[32 * 4];
  const int tid  = threadIdx.x;
  const int lane = tid & 31;
  const int wv   = tid >> 5;                 // 0..7
  const int cb0  = blockIdx.x * 32;          // column base

  // wave 0 stages the 32 column boxes into LDS (async 16B per lane)
  if (tid < 32) {
    int col  = cb0 + lane;
    int ccol = (col < PRE_NMS) ? col : (PRE_NMS - 1);   // clamp (dup harmless)
    async_copy_b128((uint32_t)(uintptr_t)(&cb[lane * 4]),
                    (unsigned long long)(uintptr_t)(bbox_s + (size_t)ccol * 4));
    async_wait0();
  }
  __syncthreads();

  const int rg = blockIdx.y * 8 + wv;        // row group 0..191
  if (rg >= MASK_WORDS) return;              // wave-uniform guard
  const int r = rg * 32 + lane;              // < ROWS_PAD

  float rx0 = bbox_s[r * 4 + 0], ry0 = bbox_s[r * 4 + 1];
  float rx1 = bbox_s[r * 4 + 2], ry1 = bbox_s[r * 4 + 3];
  float ra  = (rx1 - rx0) * (ry1 - ry0);

  uint32_t bits = 0u;
  if (cb0 + 31 > r) {                        // some j > r exists in this tile
#pragma unroll 4
    for (int c = 0; c < 32; ++c) {
      int j = cb0 + c;
      float bx0 = cb[c * 4 + 0], by0 = cb[c * 4 + 1];
      float bx1 = cb[c * 4 + 2], by1 = cb[c * 4 + 3];
      float ba  = (bx1 - bx0) * (by1 - by0);
      float xx1 = fmaxf(rx0, bx0), yy1 = fmaxf(ry0, by0);
      float xx2 = fminf(rx1, bx1), yy2 = fminf(ry1, by1);
      float inter = fmaxf(xx2 - xx1, 0.0f) * fmaxf(yy2 - yy1, 0.0f);
      float iou = inter / (ra + ba - inter + 1e-9f);
      if ((iou > NMS_THRESH) && (j > r)) bits |= (1u << c);
    }
  }
  mask[(size_t)r * MASK_WORDS + blockIdx.x] = bits;
}

// ---------------------------------------------------------------------
// 5) sequential greedy NMS scan: single wave, LDS-resident bitmaps.
//    Validity pre-packed into LDS words via wave32 ballot.
// ---------------------------------------------------------------------
__global__ void __launch_bounds__(32) k_scan(const uint32_t* __restrict__ mask,
                                             const uint32_t* __restrict__ valid_s,
                                             uint32_t* __restrict__ keep_words) {
  __shared__ uint32_t remv[MASK_WORDS];
  __shared__ uint32_t keep[MASK_WORDS];
  __shared__ uint32_t vw[MASK_WORDS];
  const int lane = threadIdx.x;

  for (int w = lane; w < MASK_WORDS; w += 32) { remv[w] = 0u; keep[w] = 0u; }
  for (int w = 0; w < MASK_WORDS; ++w) {        // pack validity bits (ballot)
    bool v = valid_s[w * 32 + lane] != 0u;
    uint32_t b = (uint32_t)__ballot(v);
    if (lane == 0) vw[w] = b;
  }
  __syncthreads();

  for (int i = 0; i < PRE_NMS; ++i) {
    const int wd = i >> 5, bt = i & 31;
    uint32_t live = vw[wd] & ~remv[wd];                    // uniform LDS reads
    bool kept = ((live >> bt) & 1u) != 0u;
    if (kept) {                                            // uniform branch
      if (lane == 0) keep[wd] |= (1u << bt);
      const uint32_t* row = mask + (size_t)i * MASK_WORDS;
      for (int w = lane; w < MASK_WORDS; w += 32) remv[w] |= row[w];
    }
    __syncthreads();
  }
  for (int w = lane; w < MASK_WORDS; w += 32) keep_words[w] = keep[w];
}

// ---------------------------------------------------------------------
// 6) stable compaction of kept boxes to front, zero-padded tail
// ---------------------------------------------------------------------
__global__ void __launch_bounds__(32) k_compact(const uint32_t* __restrict__ keep_words,
                                                const float* __restrict__ bbox_s,
                                                float* __restrict__ out) {
  const int lane = threadIdx.x;
  for (int t = lane; t < POST_NMS * 4; t += 32) out[t] = 0.0f;
  __syncthreads();

  int base = 0;
  for (int wi = 0; wi < MASK_WORDS; ++wi) {
    uint32_t word = keep_words[wi];
    int i = wi * 32 + lane;
    bool bit = ((word >> lane) & 1u) != 0u;
    int pos = base + __popc(word & ((1u << lane) - 1u));
    if (bit && pos < POST_NMS && i < PRE_NMS) {
      out[pos * 4 + 0] = bbox_s[i * 4 + 0];
      out[pos * 4 + 1] = bbox_s[i * 4 + 1];
      out[pos * 4 + 2] = bbox_s[i * 4 + 2];
      out[pos * 4 + 3] = bbox_s[i * 4 + 3];
    }
    base += __popc(word);
    if (base >= POST_NMS) break;   // deterministic early exit
  }
}

// ---------------------------------------------------------------------
extern "C" void kernel_launch(void* const* d_in, const int* in_sizes, int n_in,
                              void* d_out, int out_size, void* d_ws, size_t ws_size,
                              hipStream_t stream) {
  const float* anchors = (const float*)d_in[0];
  const float* cls     = (const float*)d_in[1];
  const float* reg     = (const float*)d_in[2];
  const int*   img_w   = (const int*)d_in[3];
  const int*   img_h   = (const int*)d_in[4];
  float*       out     = (float*)d_out;

  char* ws = (char*)d_ws;
  float*              bbox       = (float*)(ws + OFF_BBOX);
  unsigned long long* keys       = (unsigned long long*)(ws + OFF_KEYS);
  float*              bbox_s     = (float*)(ws + OFF_BBOXS);
  uint32_t*           valid_s    = (uint32_t*)(ws + OFF_VALID);
  uint32_t*           maskbuf    = (uint32_t*)(ws + OFF_MASK);
  uint32_t*           keep_words = (uint32_t*)(ws + OFF_KEEP);

  // 1) decode + keys
  k_decode<<<NSORT / 256, 256, 0, stream>>>(anchors, cls, reg, img_w, img_h,
                                            bbox, keys);

  // 2) bitonic sort, launch-minimized: 1 local-sort + 15 global + 5 local-merge
  k_sort_local<<<NSORT / SEG, 1024, 0, stream>>>(keys);
  for (int k = 2 * SEG; k <= NSORT; k <<= 1) {
    for (int j = k >> 1; j >= SEG; j >>= 1)
      k_bitonic_global<<<NSORT / 256, 256, 0, stream>>>(keys, k, j);
    k_merge_local<<<NSORT / SEG, 1024, 0, stream>>>(keys, k);
  }

  // 3) gather sorted top PRE_NMS (padded to 6016)
  k_gather<<<(ROWS_PAD + 255) / 256, 256, 0, stream>>>(keys, bbox, bbox_s,
                                                       valid_s);

  // 4) IoU bitmask: 188 x 24 blocks, 8 waves each share one column tile
  dim3 mg(MASK_WORDS, (MASK_WORDS + 7) / 8);
  k_mask<<<mg, 256, 0, stream>>>(bbox_s, maskbuf);

  // 5) greedy scan (single wave)
  k_scan<<<1, 32, 0, stream>>>(maskbuf, valid_s, keep_words);

  // 6) stable compaction -> 300x4 output
  k_compact<<<1, 32, 0, stream>>>(keep_words, bbox_s, out);
}